// InformerModel_11579231830305
// MI455X (gfx1250) — compile-verified
//
#include <hip/hip_runtime.h>
#include <hip/hip_bf16.h>

// Model dims (fixed by reference)
#define BATCH   2
#define S_LEN   2048
#define IN_DIM  64
#define DMODEL  512
#define NHEAD   8
#define DK      64
#define NLAYER  2
#define KSEL    204              // S // 10
#define ROWS    (BATCH * S_LEN)  // 4096

#define CPAD    34               // padded col stride in transposed W panel

typedef __attribute__((ext_vector_type(2))) float v2f;
typedef __attribute__((ext_vector_type(8))) float v8f;

__device__ __forceinline__ v8f wmma4(v2f a, v2f b, v8f c) {
  // D = A(16x4,f32) * B(4x16,f32) + C(16x16,f32)
  return __builtin_amdgcn_wmma_f32_16x16x4_f32(
      /*neg_a=*/false, a, /*neg_b=*/false, b,
      /*c_mod=*/(short)0, c, /*reuse_a=*/false, /*reuse_b=*/false);
}

// order-preserving float<->uint mapping (monotone: f0<f1 <=> u0<u1)
__device__ __forceinline__ unsigned f2ord(float f) {
  unsigned u = __float_as_uint(f);
  return (u & 0x80000000u) ? ~u : (u | 0x80000000u);
}
__device__ __forceinline__ float ord2f(unsigned u) {
  unsigned v = (u & 0x80000000u) ? (u & 0x7FFFFFFFu) : ~u;
  return __uint_as_float(v);
}

// ---------------------------------------------------------------------------
// C[M,N] = A[M,K] @ W[K,N] + bias (+ residual), f32 WMMA 16x16x4.
// Block (8 waves) computes a 128x64 tile. The shared 64-wide W panel is
// staged TRANSPOSED into LDS (buf[col*34 + k]) via the scatter capability of
// GLOBAL_LOAD_ASYNC_TO_LDS_B32 (per-lane LDS dest addresses), double-buffered
// on ASYNCcnt. Each WMMA B fragment is then one aligned ds_load_b64 — no
// repacking movs. Requires K % 32 == 0, M % 128 == 0, N % 64 == 0.
// ---------------------------------------------------------------------------
__global__ void gemm_wmma_f32(const float* __restrict__ A,
                              const float* __restrict__ W,
                              const float* __restrict__ bias,
                              const float* __restrict__ res,
                              float* __restrict__ C,
                              int M, int N, int K)
{
  __shared__ float wbuf[2][64 * CPAD];   // transposed, padded panel (~17 KB)

  const int tid  = threadIdx.x;
  const int lane = tid & 31;
  const int w    = tid >> 5;

  const int tiles_n = N >> 6;
  const int block_m = blockIdx.x / tiles_n;
  const int tile_n  = blockIdx.x % tiles_n;
  const int n0 = tile_n * 64;
  const int m0 = block_m * 128 + w * 16;

  const int am = lane & 15;              // A row / B col within 16x16 tile
  const int ak = (lane >> 4) * 2;        // K sub-offset for this half-wave

  // async-scatter one 32(K) x 64(N) chunk of W into wbuf[bufsel], transposed.
  // 8 x b32 per thread: global reads coalesced (128B runs), LDS writes hit
  // distinct even banks (stride 34 floats, gcd(34,64)=2).
  auto issue_chunk = [&](int kc, int bufsel) {
    #pragma unroll
    for (int s = 0; s < 8; ++s) {
      const int e   = s * 256 + tid;     // 0..2047
      const int row = e >> 6;            // K-row 0..31
      const int col = e & 63;            // N-col 0..63
      const float* gp = W + (size_t)(kc + row) * N + n0 + col;
      unsigned ldsOff = (unsigned)(uintptr_t)(&wbuf[bufsel][col * CPAD + row]);
      asm volatile("global_load_async_to_lds_b32 %0, %1, off"
                   :: "v"(ldsOff), "v"(gp) : "memory");
    }
  };

  v8f acc0 = {}, acc1 = {}, acc2 = {}, acc3 = {};
  const float* Arow = A + (size_t)(m0 + am) * K + ak;

  const int nchunks = K >> 5;
  issue_chunk(0, 0);
  for (int c = 0; c < nchunks; ++c) {
    if (c + 1 < nchunks) {
      issue_chunk((c + 1) << 5, (c + 1) & 1);
      asm volatile("s_wait_asynccnt 0x8" ::: "memory");  // chunk c landed
    } else {
      asm volatile("s_wait_asynccnt 0x0" ::: "memory");
    }
    __syncthreads();

    const float* buf = wbuf[c & 1];
    const int kc = c << 5;
    __builtin_prefetch(Arow + kc + 32, 0, 1);            // next A chunk
    #pragma unroll
    for (int kloc = 0; kloc < 32; kloc += 4) {
      v2f a;
      a.x = Arow[kc + kloc];
      a.y = Arow[kc + kloc + 1];
      const float* bp = buf + am * CPAD + kloc + ak;     // 8B-aligned
      v2f b0 = *(const v2f*)(bp);                        // ds_load_b64
      v2f b1 = *(const v2f*)(bp + 16 * CPAD);
      v2f b2 = *(const v2f*)(bp + 32 * CPAD);
      v2f b3 = *(const v2f*)(bp + 48 * CPAD);
      acc0 = wmma4(a, b0, acc0);
      acc1 = wmma4(a, b1, acc1);
      acc2 = wmma4(a, b2, acc2);
      acc3 = wmma4(a, b3, acc3);
    }
    __syncthreads();
  }

  const int mb = (lane >> 4) * 8;        // C rows: r + 8*(lane>=16)
  #pragma unroll
  for (int r = 0; r < 8; ++r) {
    const int m = m0 + mb + r;
    float* Crow = C + (size_t)m * N + n0;
    float v0 = acc0[r] + bias[n0 + am];
    float v1 = acc1[r] + bias[n0 + 16 + am];
    float v2 = acc2[r] + bias[n0 + 32 + am];
    float v3 = acc3[r] + bias[n0 + 48 + am];
    if (res) {
      const float* Rrow = res + (size_t)m * N + n0;
      v0 += Rrow[am]; v1 += Rrow[16 + am]; v2 += Rrow[32 + am]; v3 += Rrow[48 + am];
    }
    Crow[am] = v0; Crow[16 + am] = v1; Crow[32 + am] = v2; Crow[48 + am] = v3;
  }
}

// ---------------------------------------------------------------------------
// ProbSparse attention, one (b,h,16-query tile) per 256-thread workgroup.
//  1) scores(16x2048) = Q Kt / 8 via WMMA, stored in LDS as ordered uint32
//  2) exact k-th largest per row via 32-step integer bisection (wave per 2 rows)
//  3) masked softmax in place, then O = P V via WMMA (8 waves split K in half)
// LDS: 4KB Qtile + 128KB scores + 8KB partials + stats  (~140KB dynamic)
// ---------------------------------------------------------------------------
__global__ void attn_probsparse(const float* __restrict__ Q,
                                const float* __restrict__ Km,
                                const float* __restrict__ Vm,
                                float* __restrict__ O)
{
  extern __shared__ float smem[];
  float*    Qs     = smem;                                  // 16*64
  unsigned* Su     = (unsigned*)(smem + 16 * DK);           // 16*2048
  float*    part   = smem + 16 * DK + 16 * S_LEN;           // 2*16*64
  float*    rowsum = part + 2 * 16 * DK;                    // 16

  const int tid  = threadIdx.x;
  const int lane = tid & 31;
  const int w    = tid >> 5;

  const int id = blockIdx.x;
  const int qb = id & 127;            // S/16 = 128 query tiles
  const int h  = (id >> 7) & 7;
  const int b  = id >> 10;
  const int q0 = qb * 16;
  const size_t baseBH = (size_t)b * S_LEN * DMODEL + (size_t)h * DK;

  // stage Q tile (16 x 64) into LDS
  for (int e = tid; e < 16 * DK; e += 256) {
    int m = e >> 6, kk = e & 63;
    Qs[e] = Q[baseBH + (size_t)(q0 + m) * DMODEL + kk];
  }
  __syncthreads();

  const int am = lane & 15;
  const int ak = (lane >> 4) * 2;

  // ---- pass 1: scores -> ordered uints in LDS -------------------------------
  for (int kb = w; kb < S_LEN / 16; kb += 8) {
    const int key0 = kb * 16;
    v8f acc = {};
    const float* Kp = Km + baseBH + (size_t)(key0 + am) * DMODEL + ak;
    #pragma unroll
    for (int k = 0; k < DK; k += 4) {
      v2f a;  a.x = Qs[am * DK + ak + k];  a.y = Qs[am * DK + ak + k + 1];
      v2f bb; bb.x = Kp[k];                bb.y = Kp[k + 1];
      acc = wmma4(a, bb, acc);
    }
    const int mb = (lane >> 4) * 8;
    #pragma unroll
    for (int r = 0; r < 8; ++r)
      Su[(mb + r) * S_LEN + key0 + am] = f2ord(acc[r] * 0.125f);
  }
  __syncthreads();

  // ---- pass 2: exact top-k threshold + masked softmax (wave handles 2 rows) -
  for (int rr = 0; rr < 2; ++rr) {
    const int row = w * 2 + rr;
    unsigned* Sr = Su + row * S_LEN;

    unsigned um = 0;
    for (int i = lane; i < S_LEN; i += 32) { unsigned u = Sr[i]; um = u > um ? u : um; }
    #pragma unroll
    for (int off = 16; off; off >>= 1) {
      unsigned o = (unsigned)__shfl_xor((int)um, off, 32);
      um = o > um ? o : um;
    }

    unsigned lo = 0, hi = 0xFFFFFFFFu;   // invariant: cnt(>=lo)>=k, cnt(>=hi)<k
    for (int it = 0; it < 32; ++it) {
      unsigned mid = lo + ((hi - lo) >> 1);
      int c = 0;
      for (int i = lane; i < S_LEN; i += 32) c += (Sr[i] >= mid) ? 1 : 0;
      #pragma unroll
      for (int off = 16; off; off >>= 1) c += __shfl_xor(c, off, 32);
      if (c >= KSEL) lo = mid; else hi = mid;
    }

    const float mx = ord2f(um);
    float s = 0.f;
    for (int i = lane; i < S_LEN; i += 32) {
      unsigned u = Sr[i];
      float p = (u >= lo) ? __expf(ord2f(u) - mx) : 0.0f;  // score<thr -> -inf
      ((float*)Sr)[i] = p;                                  // P in place
      s += p;
    }
    #pragma unroll
    for (int off = 16; off; off >>= 1) s += __shfl_xor(s, off, 32);
    if (lane == 0) rowsum[row] = s;
  }
  __syncthreads();

  // ---- pass 3: O = P V via WMMA; waves 0-3 keys [0,1024), 4-7 keys [1024,2048)
  const float* P  = (const float*)Su;
  const int half  = w >> 2;
  const int n0    = (w & 3) * 16;
  const int k0s   = half * (S_LEN / 2);
  v8f acc = {};
  const size_t vbase = baseBH + n0 + am;
  for (int k0 = k0s; k0 < k0s + S_LEN / 2; k0 += 4) {
    v2f a;  a.x = P[am * S_LEN + k0 + ak];  a.y = P[am * S_LEN + k0 + ak + 1];
    const float* vp = Vm + vbase + (size_t)(k0 + ak) * DMODEL;
    v2f bb; bb.x = vp[0]; bb.y = vp[DMODEL];
    acc = wmma4(a, bb, acc);
  }
  {
    const int mb = (lane >> 4) * 8;
    float* pp = part + half * (16 * DK);
    #pragma unroll
    for (int r = 0; r < 8; ++r)
      pp[(mb + r) * DK + n0 + am] = acc[r];
  }
  __syncthreads();

  for (int e = tid; e < 16 * DK; e += 256) {
    int m = e >> 6, n = e & 63;
    float val = (part[e] + part[16 * DK + e]) / rowsum[m];
    O[baseBH + (size_t)(q0 + m) * DMODEL + n] = val;
  }
}

// ---------------------------------------------------------------------------
// LayerNorm over rows of 512, one wave per row (wave32: 16 elems/lane).
// ---------------------------------------------------------------------------
__global__ void layernorm_rows(const float* __restrict__ X,
                               const float* __restrict__ g,
                               const float* __restrict__ bta,
                               float* __restrict__ Y, int rows)
{
  const int lane = threadIdx.x & 31;
  const int w    = threadIdx.x >> 5;
  const int row  = blockIdx.x * 8 + w;
  if (row >= rows) return;
  const float* xr = X + (size_t)row * DMODEL;

  float v[16]; float s = 0.f;
  #pragma unroll
  for (int i = 0; i < 16; ++i) { v[i] = xr[lane + 32 * i]; s += v[i]; }
  #pragma unroll
  for (int off = 16; off; off >>= 1) s += __shfl_xor(s, off, 32);
  const float mu = s * (1.0f / DMODEL);

  float s2 = 0.f;
  #pragma unroll
  for (int i = 0; i < 16; ++i) { float d = v[i] - mu; s2 += d * d; }
  #pragma unroll
  for (int off = 16; off; off >>= 1) s2 += __shfl_xor(s2, off, 32);
  const float rstd = rsqrtf(s2 * (1.0f / DMODEL) + 1e-5f);

  float* yr = Y + (size_t)row * DMODEL;
  #pragma unroll
  for (int i = 0; i < 16; ++i) {
    int c = lane + 32 * i;
    yr[c] = (v[i] - mu) * rstd * g[c] + bta[c];
  }
}

// Decoder: (2,64) = h[:, -1, :] @ dec_W + dec_b  (tiny; plain dot products)
__global__ void decoder_last(const float* __restrict__ Hf,
                             const float* __restrict__ Wd,
                             const float* __restrict__ bd,
                             float* __restrict__ out)
{
  const int t = threadIdx.x;                 // 128 threads
  const int b = t >> 6, n = t & 63;
  const float* hr = Hf + ((size_t)b * S_LEN + (S_LEN - 1)) * DMODEL;
  float s = bd[n];
  for (int k = 0; k < DMODEL; ++k) s += hr[k] * Wd[k * IN_DIM + n];
  out[t] = s;
}

// ---------------------------------------------------------------------------
extern "C" void kernel_launch(void* const* d_in, const int* in_sizes, int n_in,
                              void* d_out, int out_size, void* d_ws, size_t ws_size,
                              hipStream_t stream)
{
  const float* x     = (const float*)d_in[0];
  const float* enc_W = (const float*)d_in[1];
  const float* enc_b = (const float*)d_in[2];
  const float* Wq    = (const float*)d_in[3];
  const float* bq    = (const float*)d_in[4];
  const float* Wk    = (const float*)d_in[5];
  const float* bk    = (const float*)d_in[6];
  const float* Wv    = (const float*)d_in[7];
  const float* bv    = (const float*)d_in[8];
  const float* fcW   = (const float*)d_in[9];
  const float* fcb   = (const float*)d_in[10];
  const float* ln_g  = (const float*)d_in[11];
  const float* ln_b  = (const float*)d_in[12];
  const float* dec_W = (const float*)d_in[13];
  const float* dec_b = (const float*)d_in[14];

  float* ws = (float*)d_ws;
  const size_t HS = (size_t)ROWS * DMODEL;   // 2,097,152 floats = 8 MB
  float* hA = ws;
  float* hB = ws + HS;
  float* Qb = ws + 2 * HS;
  float* Kb = ws + 3 * HS;
  float* Vb = ws + 4 * HS;
  float* Ab = ws + 5 * HS;

  const dim3 blk(256);
  const int gemm_blocks = (ROWS / 128) * (DMODEL / 64);      // 256
  const size_t attn_smem =
      (size_t)(16 * DK + 16 * S_LEN + 2 * 16 * DK + 16) * sizeof(float);
  const size_t DD = (size_t)DMODEL * DMODEL;

  // encoder: h = x @ enc_W + enc_b    (4096 x 64 x 512)
  gemm_wmma_f32<<<gemm_blocks, blk, 0, stream>>>(x, enc_W, enc_b, nullptr, hA,
                                                 ROWS, DMODEL, IN_DIM);

  for (int l = 0; l < NLAYER; ++l) {
    gemm_wmma_f32<<<gemm_blocks, blk, 0, stream>>>(hA, Wq + l * DD, bq + l * DMODEL,
                                                   nullptr, Qb, ROWS, DMODEL, DMODEL);
    gemm_wmma_f32<<<gemm_blocks, blk, 0, stream>>>(hA, Wk + l * DD, bk + l * DMODEL,
                                                   nullptr, Kb, ROWS, DMODEL, DMODEL);
    gemm_wmma_f32<<<gemm_blocks, blk, 0, stream>>>(hA, Wv + l * DD, bv + l * DMODEL,
                                                   nullptr, Vb, ROWS, DMODEL, DMODEL);

    attn_probsparse<<<BATCH * NHEAD * (S_LEN / 16), blk, attn_smem, stream>>>(
        Qb, Kb, Vb, Ab);

    // pre-LN: hB = attn_out @ fcW + fcb + hA   (residual fused)
    gemm_wmma_f32<<<gemm_blocks, blk, 0, stream>>>(Ab, fcW + l * DD, fcb + l * DMODEL,
                                                   hA, hB, ROWS, DMODEL, DMODEL);
    layernorm_rows<<<ROWS / 8, blk, 0, stream>>>(hB, ln_g + l * DMODEL,
                                                 ln_b + l * DMODEL, hA, ROWS);
  }

  decoder_last<<<1, 128, 0, stream>>>(hA, dec_W, dec_b, (float*)d_out);
}